// SimpleCA_75333726372311
// MI455X (gfx1250) — compile-verified
//
#include <hip/hip_runtime.h>

// NCA step for MI455X (gfx1250, wave32):
//   perception: fixed depthwise 3x3 (ident/sobelx/sobely/laplacian), circular pad
//   MLP 16->8(relu)->4 via V_WMMA_F32_16X16X4_F32 chains
//   staging: GLOBAL_LOAD_ASYNC_TO_LDS_B128 (ASYNCcnt) instead of VGPR round-trip
//   out = x + dx * floor(noise + 0.5)

typedef __attribute__((ext_vector_type(2))) float v2f;
typedef __attribute__((ext_vector_type(8))) float v8f;

#define NB  16
#define NCH 4
#define NH  512
#define NW  512
#define HID 8

__device__ __forceinline__ float swz16(float v) {
    // ds_swizzle SWAPX16: lane <-> lane^16 (group-of-32: xor=0x10, and=0x1f)
    int i = __builtin_bit_cast(int, v);
    i = __builtin_amdgcn_ds_swizzle(i, 0x401F);
    return __builtin_bit_cast(float, i);
}

__device__ __forceinline__ void wait_async0() {
#if __has_builtin(__builtin_amdgcn_s_wait_asynccnt)
    __builtin_amdgcn_s_wait_asynccnt(0);
#else
    asm volatile("s_wait_asynccnt 0x0" ::: "memory");
#endif
}

__global__ __launch_bounds__(256) void nca_step_kernel(
    const float* __restrict__ x,
    const float* __restrict__ w1w,   // [8,16]
    const float* __restrict__ w1b,   // [8]
    const float* __restrict__ w2w,   // [4,8]
    const float* __restrict__ noise, // [16,1,512,512]
    float* __restrict__ out)         // [16,4,512,512]
{
    __shared__ __align__(16) float ls[3][NCH][NW];   // 24 KB: rows h-1,h,h+1 x 4 ch

    const int bid = blockIdx.x;            // 0 .. NB*NH-1
    const int b   = bid >> 9;
    const int h   = bid & (NH - 1);
    const int hm  = (h + NH - 1) & (NH - 1);
    const int hp  = (h + 1) & (NH - 1);
    const int t   = threadIdx.x;

    // ---- stage 3 wrapped rows x 4 channels into LDS via async global->LDS ----
    // 3*4*512 floats = 1536 x b128 -> 6 per thread; tracked by ASYNCcnt.
    // Generic->LDS address: low 32 bits of generic pointer (ISA aperture rule).
    const uint32_t lds_base = (uint32_t)(uintptr_t)&ls[0][0][0];
    #pragma unroll
    for (int i = 0; i < 6; ++i) {
        int idx4 = t + i * 256;            // 0..1535 (float4 granules)
        int rr   = idx4 >> 7;              // 0..11  (row-chan pair)
        int c4   = idx4 & 127;             // float4 index within row
        int r3   = rr >> 2;                // 0..2
        int c    = rr & 3;                 // channel
        int grow = (r3 == 0) ? hm : ((r3 == 1) ? h : hp);
        uint32_t goff = (uint32_t)(((((b * NCH + c) * NH + grow) * NW) + (c4 << 2)) * 4);
        uint32_t ldsa = lds_base + (uint32_t)(idx4 << 4);
        // GVS mode: saddr(u64, uniform) + vaddr(i32 byte offset)
        asm volatile("global_load_async_to_lds_b128 %0, %1, %2 offset:0"
                     :: "v"(ldsa), "v"(goff), "s"(x) : "memory");
    }
    wait_async0();
    __syncthreads();

    const int lane = t & 31;
    const int wv   = t >> 5;               // wave 0..7, each owns 64 pixels
    const int rl   = lane & 15;
    const bool lo  = lane < 16;
    const int koff = lo ? 0 : 2;           // K offset for upper half-wave (ISA f32 layout)

    // ---- W1 in A-matrix layout: chunk c covers K=4c..4c+3 ----
    v2f a1[4];
    #pragma unroll
    for (int c = 0; c < 4; ++c) {
        float e0 = 0.0f, e1 = 0.0f;
        if (rl < HID) {
            e0 = w1w[rl * 16 + 4 * c + koff];
            e1 = w1w[rl * 16 + 4 * c + 1 + koff];
        }
        a1[c][0] = e0; a1[c][1] = e1;
    }
    // ---- W2 in A-matrix layout: 2 chunks over K=0..7 ----
    v2f a2[2];
    #pragma unroll
    for (int c = 0; c < 2; ++c) {
        float e0 = 0.0f, e1 = 0.0f;
        if (rl < NCH) {
            e0 = w2w[rl * 8 + 4 * c + koff];
            e1 = w2w[rl * 8 + 4 * c + 1 + koff];
        }
        a2[c][0] = e0; a2[c][1] = e1;
    }
    float bias[8];
    #pragma unroll
    for (int r = 0; r < 8; ++r) bias[r] = w1b[r];

    // ---- 4 groups of 16 pixels per wave ----
    #pragma unroll 1
    for (int g = 0; g < 4; ++g) {
        const int wc = wv * 64 + g * 16 + rl;        // pixel column (lane&15)
        const int wl = (wc + NW - 1) & (NW - 1);
        const int wr = (wc + 1) & (NW - 1);

        // layer 1: acc = W1[16x16pad] x feat[16x16], K chunks of 4
        v8f acc = {};
        #pragma unroll
        for (int c = 0; c < 4; ++c) {
            float tl = ls[0][c][wl], tc = ls[0][c][wc], tr = ls[0][c][wr];
            float ml = ls[1][c][wl], mc = ls[1][c][wc], mr = ls[1][c][wr];
            float bl = ls[2][c][wl], bc = ls[2][c][wc], br = ls[2][c][wr];
            float f0 = mc;                                              // identity
            float f1 = (tr - tl) + 2.0f * (mr - ml) + (br - bl);        // sobel_x
            float f2 = (bl - tl) + 2.0f * (bc - tc) + (br - tr);        // sobel_y
            float f3 = (tl + 2.0f * tc + tr) + (2.0f * ml - 12.0f * mc + 2.0f * mr)
                     + (bl + 2.0f * bc + br);                           // laplacian
            v2f bb;
            bb[0] = lo ? f0 : f2;   // K=4c+0 / 4c+2
            bb[1] = lo ? f1 : f3;   // K=4c+1 / 4c+3
            acc = __builtin_amdgcn_wmma_f32_16x16x4_f32(
                false, a1[c], false, bb, (short)0, acc, false, false);
        }

        // bias + relu (rows 0..7 live in lanes 0-15 of acc[0..7])
        float hv[8];
        #pragma unroll
        for (int r = 0; r < 8; ++r) hv[r] = fmaxf(acc[r] + bias[r], 0.0f);

        // move hidden rows 2,3,6,7 into lanes 16-31 (full-EXEC swizzles first)
        float s2 = swz16(hv[2]), s3 = swz16(hv[3]);
        float s6 = swz16(hv[6]), s7 = swz16(hv[7]);
        v2f b20, b21;
        b20[0] = lo ? hv[0] : s2;
        b20[1] = lo ? hv[1] : s3;
        b21[0] = lo ? hv[4] : s6;
        b21[1] = lo ? hv[5] : s7;

        // layer 2: dx = W2[4x8pad] x hidden[8x16]
        v8f acc2 = {};
        acc2 = __builtin_amdgcn_wmma_f32_16x16x4_f32(
            false, a2[0], false, b20, (short)0, acc2, false, false);
        acc2 = __builtin_amdgcn_wmma_f32_16x16x4_f32(
            false, a2[1], false, b21, (short)0, acc2, false, false);

        // stochastic update + store (channel r in acc2[r], pixel = lane 0..15)
        if (lo) {
            float nz = noise[((size_t)b * NH + h) * NW + wc];
            float m  = (nz >= 0.5f) ? 1.0f : 0.0f;   // floor(U[0,1) + 0.5)
            #pragma unroll
            for (int r = 0; r < NCH; ++r) {
                float xv = ls[1][r][wc];
                out[(((size_t)b * NCH + r) * NH + h) * NW + wc] = xv + acc2[r] * m;
            }
        }
    }
}

extern "C" void kernel_launch(void* const* d_in, const int* in_sizes, int n_in,
                              void* d_out, int out_size, void* d_ws, size_t ws_size,
                              hipStream_t stream) {
    const float* x     = (const float*)d_in[0];
    const float* w1w   = (const float*)d_in[1];
    const float* w1b   = (const float*)d_in[2];
    const float* w2w   = (const float*)d_in[3];
    const float* noise = (const float*)d_in[4];
    float* out = (float*)d_out;

    dim3 grid(NB * NH);   // 8192 blocks: one (batch,row) per block
    dim3 block(256);      // 8 wave32
    nca_step_kernel<<<grid, block, 0, stream>>>(x, w1w, w1b, w2w, noise, out);
}